// DynamicRouting_51960514347592
// MI455X (gfx1250) — compile-verified
//
#include <hip/hip_runtime.h>
#include <math.h>

// ---------------------------------------------------------------------------
// Dynamic routing (capsule nets), MI455X / gfx1250.
// HBM-bound (0.5 flop/byte): minimize passes over the 302MB u_hat tensor.
//   - per-(b,j) blocks stage the full 1152x16 f32 u-tile in LDS (92KB padded;
//     WGP has 320KB), fuse s-GEMV + squash + b-update GEMV per u_hat read.
//   - sparsified capsules skip their u-tile read entirely (20/32 then 12/32):
//     total u_hat traffic = (1 + 20/32 + 12/32) * 302MB ~= 604MB ~= 26us.
//   - MAC work done with chained V_WMMA_F32_16X16X4_F32 (K=4 f32 WMMA).
//   - tile staging uses GLOBAL_LOAD_ASYNC_TO_LDS_B128 (ASYNCcnt) when the
//     toolchain exposes it; VGPR-staged float4 path otherwise.
// ---------------------------------------------------------------------------

#define BATCH 128
#define JCAPS 32
#define IDIM  1152
#define NDIM  16
#define USTRIDE 20  // padded LDS row stride (floats): bank-conflict-free AND
                    // keeps every row/chunk 16B-aligned for async B128 stores

typedef float v2f __attribute__((ext_vector_type(2)));
typedef float v8f __attribute__((ext_vector_type(8)));
typedef float f4v __attribute__((ext_vector_type(4)));
typedef int   v4i __attribute__((ext_vector_type(4)));

#if __has_builtin(__builtin_amdgcn_global_load_async_to_lds_b128) && \
    __has_builtin(__builtin_amdgcn_s_wait_asynccnt)
#define HAVE_ASYNC_LDS 1
#else
#define HAVE_ASYNC_LDS 0
#endif

__device__ __forceinline__ v8f wmma4(v2f a, v2f b, v8f c) {
  // D(16x16) = A(16x4) x B(4x16) + C, all f32
  return __builtin_amdgcn_wmma_f32_16x16x4_f32(false, a, false, b, (short)0, c,
                                               false, false);
}

// ---------------------------------------------------------------------------
// init: selmask = all ones, entropy[iter0] = log(32) (softmax of zeros is
// uniform, so entropy is exactly log J for every (b,i)).
// ---------------------------------------------------------------------------
__global__ void k_init(unsigned* __restrict__ selmask, float* __restrict__ ent) {
  int t = threadIdx.x;
  if (t < BATCH) {
    selmask[t] = 0xFFFFFFFFu;
    ent[t * 3 + 0] = logf(32.0f);
  }
}

// ---------------------------------------------------------------------------
// fused per-(b,j) pass.  MODE bits:
//   1 : c from c_buf (else uniform 1/32)
//   2 : compute b-update (write/accumulate b_buf)
//   4 : accumulate onto existing b_buf (iter >= 1)
//   8 : final iteration: write v to v_out (and zeros for masked capsules)
// ---------------------------------------------------------------------------
template <int MODE>
__global__ __launch_bounds__(256) void k_fused(
    const float* __restrict__ u, const float* __restrict__ bias,
    const float* __restrict__ c_buf, float* __restrict__ b_buf,
    float* __restrict__ v_out, const unsigned* __restrict__ selmask) {
  extern __shared__ float smem[];
  float* su = smem;                    // IDIM * USTRIDE   (16B-aligned rows)
  float* sc = su + IDIM * USTRIDE;     // IDIM
  float* sp = sc + IDIM;               // 8 waves * 16 partial s
  float* sv = sp + 128;                // 16  (v vector)
  float* ss = sv + 16;                 // 16  (s vector)

  const int blk = blockIdx.x;
  const int bb = blk >> 5, jj = blk & 31;
  const int tid = threadIdx.x;

  if (selmask) {
    if (!((selmask[bb] >> jj) & 1u)) {
      if (MODE & 8) {
        if (tid < NDIM) v_out[(size_t)(bb * JCAPS + jj) * NDIM + tid] = 0.0f;
      }
      return;  // whole block exits uniformly; masked capsule contributes v=0
    }
  }

  const size_t ubase = (size_t)(bb * JCAPS + jj) * (IDIM * NDIM);

  // ---- stage u tile (and c column) into LDS ----
#if HAVE_ASYNC_LDS
  for (int d = tid * 4; d < IDIM * NDIM; d += 256 * 4) {
    const int i = d >> 4, n0 = d & 15;
    __builtin_amdgcn_global_load_async_to_lds_b128(
        (v4i*)(u + ubase + d), (v4i*)(su + i * USTRIDE + n0), 0, 0);
  }
  if (MODE & 1) {
    const float* cc = c_buf + (size_t)(bb * JCAPS + jj) * IDIM;
    for (int d = tid * 4; d < IDIM; d += 256 * 4)
      __builtin_amdgcn_global_load_async_to_lds_b128(
          (v4i*)(cc + d), (v4i*)(sc + d), 0, 0);
  } else {
    for (int i = tid; i < IDIM; i += 256) sc[i] = 1.0f / 32.0f;
  }
  __builtin_amdgcn_s_wait_asynccnt(0);
#else
  for (int d = tid * 4; d < IDIM * NDIM; d += 256 * 4) {
    f4v t = *(const f4v*)(u + ubase + d);
    *(f4v*)(su + (d >> 4) * USTRIDE + (d & 15)) = t;  // 16B-aligned (stride 20)
  }
  if (MODE & 1) {
    const float* cc = c_buf + (size_t)(bb * JCAPS + jj) * IDIM;
    for (int i = tid; i < IDIM; i += 256) sc[i] = cc[i];
  } else {
    for (int i = tid; i < IDIM; i += 256) sc[i] = 1.0f / 32.0f;
  }
#endif
  __syncthreads();

  const int wave = tid >> 5;
  const int lane = tid & 31;
  const int lm = lane & 15;
  const bool hi = lane >= 16;

  // ---- phase 1: s[n] = sum_i c[i] * u[i,n] via WMMA ----
  // A[m,k] = u(i0+k0+k, m)  (transposed chunk), B[k,*] = c[i0+k0+k] broadcast.
  v8f acc = {0.f, 0.f, 0.f, 0.f, 0.f, 0.f, 0.f, 0.f};
  for (int q = 0; q < 9; ++q) {
    const int i0 = (wave * 9 + q) * 16;
#pragma unroll
    for (int k0 = 0; k0 < 16; k0 += 4) {
      const int kb = i0 + k0 + (hi ? 2 : 0);
      v2f a, b;
      a.x = su[(kb + 0) * USTRIDE + lm];
      a.y = su[(kb + 1) * USTRIDE + lm];
      b.x = sc[kb + 0];
      b.y = sc[kb + 1];
      acc = wmma4(a, b, acc);
    }
  }
  // D layout: VGPR r -> (M=r, lanes0-15) / (M=8+r, lanes16-31); all N equal.
  if (lane == 0) {
#pragma unroll
    for (int r = 0; r < 8; ++r) sp[wave * 16 + r] = acc[r];
  } else if (lane == 16) {
#pragma unroll
    for (int r = 0; r < 8; ++r) sp[wave * 16 + 8 + r] = acc[r];
  }
  __syncthreads();

  if (tid < NDIM) {
    float s = 0.f;
#pragma unroll
    for (int w = 0; w < 8; ++w) s += sp[w * 16 + tid];
    ss[tid] = s;
  }
  __syncthreads();

  // ---- phase 2: reset-mask + bias + squash ----
  if (tid < NDIM) {
    float sum_s = 0.f;
#pragma unroll
    for (int n = 0; n < NDIM; ++n) sum_s += ss[n];
    const bool reset = (sum_s == 0.0f);
    float sq = 0.f;
#pragma unroll
    for (int n = 0; n < NDIM; ++n) {
      float sb = reset ? 0.0f : (ss[n] + bias[jj * NDIM + n]);
      sq += sb * sb;
    }
    float mysb = reset ? 0.0f : (ss[tid] + bias[jj * NDIM + tid]);
    float scale = (sq / (1.0f + sq)) / sqrtf(sq + 1e-8f);
    float v = mysb * scale;
    sv[tid] = v;
    if (MODE & 8) v_out[(size_t)(bb * JCAPS + jj) * NDIM + tid] = v;
  }
  __syncthreads();

  // ---- phase 3: bupd[i] = sum_n u[i,n] * v[n] via WMMA ----
  if constexpr ((MODE & 2) != 0) {
    // A[m,k] = u(i0+m, n0+k), B[k,*] = v[n0+k] broadcast; D col0 = bupd.
    for (int q = 0; q < 9; ++q) {
      const int i0 = (wave * 9 + q) * 16;
      v8f accB = {0.f, 0.f, 0.f, 0.f, 0.f, 0.f, 0.f, 0.f};
#pragma unroll
      for (int n0 = 0; n0 < 16; n0 += 4) {
        const int nb = n0 + (hi ? 2 : 0);
        v2f a, b;
        a.x = su[(i0 + lm) * USTRIDE + nb + 0];
        a.y = su[(i0 + lm) * USTRIDE + nb + 1];
        b.x = sv[nb + 0];
        b.y = sv[nb + 1];
        accB = wmma4(a, b, accB);
      }
      if (lane == 0 || lane == 16) {
        float* bp = b_buf + (size_t)(bb * JCAPS + jj) * IDIM + i0 + (hi ? 8 : 0);
#pragma unroll
        for (int r = 0; r < 8; ++r) {
          float val = accB[r];
          if (MODE & 4) val += bp[r];
          bp[r] = val;
        }
      }
    }
  }
}

// ---------------------------------------------------------------------------
// column pass over b_buf: per (b,i) masked softmax over j.
//   do_scores: scores[b,j] = sum_i c / IDIM   (for top-k sparsify)
//   write c_buf + entropy slot (post-mask softmax for next iteration)
// One block per batch element; 384 threads * 3 i's each = 1152.
// ---------------------------------------------------------------------------
__global__ __launch_bounds__(384) void k_colpass(
    const float* __restrict__ b_buf, const unsigned* __restrict__ selmask,
    float* __restrict__ scores, float* __restrict__ c_buf,
    float* __restrict__ ent_base, int slot, int do_scores, int do_ent) {
  __shared__ float ssc[JCAPS];
  __shared__ float sent[384];
  const int b = blockIdx.x, tid = threadIdx.x;
  const unsigned mask = selmask[b];

  float jacc[JCAPS];
#pragma unroll
  for (int j = 0; j < JCAPS; ++j) jacc[j] = 0.f;
  float eacc = 0.f;

  for (int i = tid; i < IDIM; i += 384) {
    float val[JCAPS];
    float m = -INFINITY;
#pragma unroll
    for (int j = 0; j < JCAPS; ++j) {
      val[j] = b_buf[(size_t)(b * JCAPS + j) * IDIM + i];
      if ((mask >> j) & 1u) m = fmaxf(m, val[j]);
    }
    float denom = 0.f;
#pragma unroll
    for (int j = 0; j < JCAPS; ++j)
      if ((mask >> j) & 1u) denom += expf(val[j] - m);
    const float ld = logf(denom);
#pragma unroll
    for (int j = 0; j < JCAPS; ++j) {
      float c = ((mask >> j) & 1u) ? expf(val[j] - m) / denom : 0.0f;
      jacc[j] += c;
      if (c_buf) c_buf[(size_t)(b * JCAPS + j) * IDIM + i] = c;
      if (do_ent && c > 0.0f) eacc -= c * ((val[j] - m) - ld);
    }
  }

  if (do_scores) {
    if (tid < JCAPS) ssc[tid] = 0.f;
    __syncthreads();
#pragma unroll
    for (int j = 0; j < JCAPS; ++j) atomicAdd(&ssc[j], jacc[j]);
    __syncthreads();
    if (tid < JCAPS) scores[b * JCAPS + tid] = ssc[tid] * (1.0f / IDIM);
  }
  if (do_ent) {
    sent[tid] = eacc;
    __syncthreads();
    if (tid < 128) sent[tid] += sent[tid + 128] + sent[tid + 256];
    __syncthreads();
    for (int off = 64; off > 0; off >>= 1) {
      if (tid < off) sent[tid] += sent[tid + off];
      __syncthreads();
    }
    if (tid == 0) ent_base[b * 3 + slot] = sent[0] * (1.0f / IDIM);
  }
}

// ---------------------------------------------------------------------------
// top-k over 32 scores per batch (serial argmax: first-index wins on ties,
// matching lax.top_k). Prior-masked capsules have score exactly 0; keep <=
// count of positive scores, so the new mask is a subset of the old one.
// ---------------------------------------------------------------------------
__global__ void k_topk(const float* __restrict__ scores,
                       unsigned* __restrict__ selmask, int keep) {
  const int b = threadIdx.x;
  if (b >= BATCH) return;
  unsigned m = 0;
  for (int k = 0; k < keep; ++k) {
    float best = -INFINITY;
    int bj = 0;
    for (int j = 0; j < JCAPS; ++j) {
      float s = scores[b * JCAPS + j];
      if (!((m >> j) & 1u) && s > best) { best = s; bj = j; }
    }
    m |= 1u << bj;
  }
  selmask[b] = m;
}

// ---------------------------------------------------------------------------
extern "C" void kernel_launch(void* const* d_in, const int* in_sizes, int n_in,
                              void* d_out, int out_size, void* d_ws,
                              size_t ws_size, hipStream_t stream) {
  (void)in_sizes; (void)n_in; (void)out_size; (void)ws_size;
  const float* u = (const float*)d_in[0];      // (128,32,1152,16) f32
  const float* bias = (const float*)d_in[1];   // (32,16) f32
  // d_in[2] = iters (device scalar); schedule compiled for iters == 3.

  float* out = (float*)d_out;
  float* v_out = out;                            // 128*32*16
  float* ent = out + BATCH * JCAPS * NDIM;       // 128*3

  float* b_buf = (float*)d_ws;                                   // 4.72M f32
  float* c_buf = b_buf + (size_t)BATCH * JCAPS * IDIM;           // 4.72M f32
  float* scores = c_buf + (size_t)BATCH * JCAPS * IDIM;          // 4096 f32
  unsigned* selmask = (unsigned*)(scores + BATCH * JCAPS);       // 128 u32

  const size_t smem =
      (size_t)(IDIM * USTRIDE + IDIM + 128 + 16 + 16) * sizeof(float);
  const dim3 gF(BATCH * JCAPS), bF(256);

  k_init<<<1, 128, 0, stream>>>(selmask, ent);

  // iter 0: uniform c, fused s + squash + b-update (b1 = u.v0)
  k_fused<2><<<gF, bF, smem, stream>>>(u, bias, nullptr, b_buf, nullptr, nullptr);
  // sparsify round 0: scores (pre-mask softmax), keep 20
  k_colpass<<<BATCH, 384, 0, stream>>>(b_buf, selmask, scores, nullptr, nullptr,
                                       0, 1, 0);
  k_topk<<<1, 128, 0, stream>>>(scores, selmask, 20);
  // c1 = masked softmax(b1) + entropy slot 1
  k_colpass<<<BATCH, 384, 0, stream>>>(b_buf, selmask, scores, c_buf, ent,
                                       1, 0, 1);

  // iter 1: fused s + squash + b-update (b2 = b1 + u.v1), masked blocks skip
  k_fused<1 | 2 | 4><<<gF, bF, smem, stream>>>(u, bias, c_buf, b_buf, nullptr,
                                               selmask);
  // sparsify round 1: keep 12
  k_colpass<<<BATCH, 384, 0, stream>>>(b_buf, selmask, scores, nullptr, nullptr,
                                       0, 1, 0);
  k_topk<<<1, 128, 0, stream>>>(scores, selmask, 12);
  // c2 = masked softmax(b2) + entropy slot 2
  k_colpass<<<BATCH, 384, 0, stream>>>(b_buf, selmask, scores, c_buf, ent,
                                       2, 0, 1);

  // iter 2 (final): s + squash only, v to d_out (zeros for masked capsules)
  k_fused<1 | 8><<<gF, bF, smem, stream>>>(u, bias, c_buf, b_buf, v_out,
                                           selmask);
}